// BaseModel_78829829750856
// MI455X (gfx1250) — compile-verified
//
#include <hip/hip_runtime.h>

typedef float v2f __attribute__((ext_vector_type(2)));
typedef float v8f __attribute__((ext_vector_type(8)));

#define HID 128
#define OUTD 64

// ---------------------------------------------------------------- utilities
__global__ void fill_f32_kernel(float* __restrict__ p, long n, float v) {
    long i = (long)blockIdx.x * blockDim.x + threadIdx.x;
    if (i < n) p[i] = v;
}

__global__ void deg_count_kernel(const int* __restrict__ dst, float* __restrict__ deg, int E) {
    int e = blockIdx.x * blockDim.x + threadIdx.x;
    if (e < E) atomicAdd(&deg[dst[e]], 1.0f);
}

__global__ void deg_rsqrt_kernel(float* __restrict__ deg, int N) {
    int i = blockIdx.x * blockDim.x + threadIdx.x;
    if (i < N) deg[i] = rsqrtf(deg[i] + 1.0f);   // self-loop included
}

// ------------------------------------------------- WMMA f32 GEMM (CDNA5)
// C[M, CT*16] = A[M,K] @ B[K, CT*16] (+ bias) (optional relu).
// B is staged once per workgroup into LDS, pre-paired so each lane's B
// operand for V_WMMA_F32_16X16X4_F32 is a single contiguous ds_load_b64:
//   Bs[k2][col] = (B[2*k2][col], B[2*k2+1][col]);  pair index = k/2 + half.
// One wave32 per 16-row strip; A loaded once per K-step, reused across CT
// independent accumulator chains. M%16==0, K%4==0, N==CT*16 required.
template <int CT>
__global__ void gemm_strip_wmma_f32(const float* __restrict__ A, const float* __restrict__ B,
                                    const float* __restrict__ bias, float* __restrict__ C,
                                    int M, int K, int relu) {
    const int N = CT * 16;
    extern __shared__ float2 Bs[];             // [K/2][N]

    // ---- cooperative staging of B into LDS (coalesced: consecutive cols)
    const int tid = (int)threadIdx.x;
    const int pairs = (K >> 1) * N;
    for (int idx = tid; idx < pairs; idx += (int)blockDim.x) {
        const int k2  = idx / N;
        const int col = idx - k2 * N;
        const size_t r0 = (size_t)(2 * k2) * N + col;
        Bs[idx] = make_float2(B[r0], B[r0 + N]);
    }
    __syncthreads();                           // all waves reach this barrier

    const int strips = M >> 4;
    const int strip = blockIdx.x * ((int)blockDim.x >> 5) + (tid >> 5);
    if (strip >= strips) return;               // wave-uniform exit (post-barrier)

    const int lane = tid & 31;
    const int half = lane >> 4;                // 0: lanes 0-15, 1: lanes 16-31
    const int l16  = lane & 15;
    const int row0 = strip << 4;

    v8f acc[CT] = {};
    const float* __restrict__ arow = A + (size_t)(row0 + l16) * K;

    for (int k = 0; k < K; k += 4) {
        const int ka = k + half * 2;                       // K index of VGPR0
        const v2f a = *(const v2f*)(arow + ka);            // A[m][ka], A[m][ka+1]
        const float2* __restrict__ brow = Bs + (size_t)((k >> 1) + half) * N + l16;
        #pragma unroll
        for (int t = 0; t < CT; ++t) {
            const float2 bp = brow[t * 16];                // one ds_load_b64
            v2f b; b.x = bp.x; b.y = bp.y;                 // B[ka][n], B[ka+1][n]
            acc[t] = __builtin_amdgcn_wmma_f32_16x16x4_f32(
                /*neg_a=*/false, a, /*neg_b=*/false, b,
                /*c_mod=*/(short)0, acc[t], /*reuse_a=*/false, /*reuse_b=*/false);
        }
    }

    #pragma unroll
    for (int t = 0; t < CT; ++t) {
        const int col = t * 16 + l16;
        const float bv = bias ? bias[col] : 0.0f;
        #pragma unroll
        for (int r = 0; r < 8; ++r) {
            const int row = row0 + r + half * 8;           // C/D: VGPR r -> M=r / r+8
            float v = acc[t][r] + bv;
            if (relu) v = fmaxf(v, 0.0f);
            C[(size_t)row * N + col] = v;
        }
    }
}

// --------------------------------------------- edge gather + scatter-add
// One thread per (edge, 4-feature chunk): 32 threads cover 128 features.
__global__ void edge_scatter_kernel(const int* __restrict__ src, const int* __restrict__ dst,
                                    const float* __restrict__ dis, const float* __restrict__ hW,
                                    float* __restrict__ agg, int E) {
    long t = (long)blockIdx.x * blockDim.x + threadIdx.x;
    int e = (int)(t >> 5);
    if (e >= E) return;
    int part = (int)t & 31;
    int s = src[e], d = dst[e];
    float norm = dis[s] * dis[d];
    float4 v = ((const float4*)(hW + (size_t)s * HID))[part];
    float* o = agg + (size_t)d * HID + part * 4;
    atomicAdd(o + 0, v.x * norm);
    atomicAdd(o + 1, v.y * norm);
    atomicAdd(o + 2, v.z * norm);
    atomicAdd(o + 3, v.w * norm);
}

// agg + hW * dis^2 (self loop) + bias, optional relu
__global__ void gcn_combine_kernel(const float* __restrict__ agg, const float* __restrict__ hW,
                                   const float* __restrict__ dis, const float* __restrict__ b,
                                   float* __restrict__ out, int N, int relu) {
    long t = (long)blockIdx.x * blockDim.x + threadIdx.x;
    if (t >= (long)N * HID) return;
    int i = (int)(t >> 7);                     // /128
    int f = (int)t & 127;
    float di = dis[i];
    float v = agg[t] + hW[t] * (di * di) + b[f];
    if (relu) v = fmaxf(v, 0.0f);
    out[t] = v;
}

// ---------------------------------------------------- generalized-mean pool
__global__ void pool_accum_kernel(const float* __restrict__ h, const int* __restrict__ batch,
                                  float* __restrict__ sums, float* __restrict__ cnt, int N) {
    long t = (long)blockIdx.x * blockDim.x + threadIdx.x;
    int i = (int)(t >> 5);
    if (i >= N) return;
    int part = (int)t & 31;
    int g = batch[i];
    float4 v = ((const float4*)(h + (size_t)i * HID))[part];
    float* o = sums + (size_t)g * HID + part * 4;
    // xt = sign(x) * (|x| + eps)^2  (p = 2)
    #define GMXT(x) ((x) == 0.0f ? 0.0f : ((x) > 0.0f ? (fabsf(x)+1e-6f)*(fabsf(x)+1e-6f) \
                                                      : -((fabsf(x)+1e-6f)*(fabsf(x)+1e-6f))))
    atomicAdd(o + 0, GMXT(v.x));
    atomicAdd(o + 1, GMXT(v.y));
    atomicAdd(o + 2, GMXT(v.z));
    atomicAdd(o + 3, GMXT(v.w));
    #undef GMXT
    if (part == 0) atomicAdd(&cnt[g], 1.0f);
}

// mean -> sign(m)*(|m|+eps)^(1/2) -> relu   (in place)
__global__ void pool_finalize_kernel(float* __restrict__ sums, const float* __restrict__ cnt,
                                     int G) {
    long t = (long)blockIdx.x * blockDim.x + threadIdx.x;
    if (t >= (long)G * HID) return;
    int g = (int)(t >> 7);
    float c = fmaxf(cnt[g], 1.0f);
    float m = sums[t] / c;
    float r;
    if (m > 0.0f)      r = sqrtf(m + 1e-6f);
    else if (m < 0.0f) r = -sqrtf(-m + 1e-6f);
    else               r = 0.0f;
    sums[t] = fmaxf(r, 0.0f);                  // relu fused
}

// ---------------------------------------------------------------- launcher
extern "C" void kernel_launch(void* const* d_in, const int* in_sizes, int n_in,
                              void* d_out, int out_size, void* d_ws, size_t ws_size,
                              hipStream_t stream) {
    const float* x    = (const float*)d_in[0];
    const int*   ei   = (const int*)d_in[1];
    const int*   bat  = (const int*)d_in[2];
    const float* W0   = (const float*)d_in[3];  const float* b0  = (const float*)d_in[4];
    const float* W1   = (const float*)d_in[5];  const float* b1  = (const float*)d_in[6];
    const float* W2   = (const float*)d_in[7];  const float* b2  = (const float*)d_in[8];
    const float* Wf0  = (const float*)d_in[9];  const float* bf0 = (const float*)d_in[10];
    const float* Wf1  = (const float*)d_in[11]; const float* bf1 = (const float*)d_in[12];
    float* out = (float*)d_out;

    const int N = in_sizes[2];          // 50000 nodes (multiple of 16)
    const int E = in_sizes[1] / 2;      // 800000 edges
    const int G = out_size / OUTD;      // 512 graphs

    const int* src = ei;
    const int* dst = ei + E;

    // workspace layout
    char* ws = (char*)d_ws;
    const size_t SZ = (size_t)N * HID * sizeof(float);
    float* bufA   = (float*)(ws);
    float* bufB   = (float*)(ws + SZ);
    size_t off    = 2 * SZ;
    float* deg    = (float*)(ws + off);  off += ((size_t)N * 4 + 255) & ~(size_t)255;
    float* pooled = (float*)(ws + off);  off += ((size_t)G * HID * 4 + 255) & ~(size_t)255;
    float* cnt    = (float*)(ws + off);  off += ((size_t)G * 4 + 255) & ~(size_t)255;
    float* zbuf   = (float*)(ws + off);

    const int T = 256;
    const long NH = (long)N * HID;
    dim3 blkNH((unsigned)((NH + T - 1) / T));
    dim3 blkE32((unsigned)(((long)E * 32 + T - 1) / T));
    dim3 blkN32((unsigned)(((long)N * 32 + T - 1) / T));

    // dynamic LDS for staged weights: (K/2) * N * sizeof(float2)
    const size_t shB8 = (size_t)(HID / 2) * 128 * sizeof(float2);   // 64 KB
    const size_t shB4 = (size_t)(HID / 2) * 64  * sizeof(float2);   // 32 KB

    // degrees (with self loop) -> deg_inv_sqrt
    fill_f32_kernel<<<dim3((N + T - 1) / T), T, 0, stream>>>(deg, N, 0.0f);
    deg_count_kernel<<<dim3((E + T - 1) / T), T, 0, stream>>>(dst, deg, E);
    deg_rsqrt_kernel<<<dim3((N + T - 1) / T), T, 0, stream>>>(deg, N);

    const int strips = N >> 4;                         // 16-row strips, 4 waves/block
    dim3 gBlocks((strips + 3) / 4);

    // ---- layer 0: hin=x, hW->bufA, agg=bufB, out->bufA, relu
    gemm_strip_wmma_f32<8><<<gBlocks, 128, shB8, stream>>>(x, W0, nullptr, bufA, N, HID, 0);
    fill_f32_kernel<<<blkNH, T, 0, stream>>>(bufB, NH, 0.0f);
    edge_scatter_kernel<<<blkE32, T, 0, stream>>>(src, dst, deg, bufA, bufB, E);
    gcn_combine_kernel<<<blkNH, T, 0, stream>>>(bufB, bufA, deg, b0, bufA, N, 1);

    // ---- layer 1: hin=bufA, hW->bufB, agg=bufA, out->bufB, relu
    gemm_strip_wmma_f32<8><<<gBlocks, 128, shB8, stream>>>(bufA, W1, nullptr, bufB, N, HID, 0);
    fill_f32_kernel<<<blkNH, T, 0, stream>>>(bufA, NH, 0.0f);
    edge_scatter_kernel<<<blkE32, T, 0, stream>>>(src, dst, deg, bufB, bufA, E);
    gcn_combine_kernel<<<blkNH, T, 0, stream>>>(bufA, bufB, deg, b1, bufB, N, 1);

    // ---- layer 2: hin=bufB, hW->bufA, agg=bufB, out->bufA, no relu
    gemm_strip_wmma_f32<8><<<gBlocks, 128, shB8, stream>>>(bufB, W2, nullptr, bufA, N, HID, 0);
    fill_f32_kernel<<<blkNH, T, 0, stream>>>(bufB, NH, 0.0f);
    edge_scatter_kernel<<<blkE32, T, 0, stream>>>(src, dst, deg, bufA, bufB, E);
    gcn_combine_kernel<<<blkNH, T, 0, stream>>>(bufB, bufA, deg, b2, bufA, N, 0);

    // ---- generalized-mean pool (p=2), relu fused into finalize
    const long GH = (long)G * HID;
    fill_f32_kernel<<<dim3((unsigned)((GH + T - 1) / T)), T, 0, stream>>>(pooled, GH, 0.0f);
    fill_f32_kernel<<<dim3((G + T - 1) / T), T, 0, stream>>>(cnt, G, 0.0f);
    pool_accum_kernel<<<blkN32, T, 0, stream>>>(bufA, bat, pooled, cnt, N);
    pool_finalize_kernel<<<dim3((unsigned)((GH + T - 1) / T)), T, 0, stream>>>(pooled, cnt, G);

    // ---- FF head: relu(pooled @ Wf0 + bf0) @ Wf1 + bf1
    const int fStrips = G >> 4;                        // 32 strips
    gemm_strip_wmma_f32<8><<<dim3((fStrips + 3) / 4), 128, shB8, stream>>>(pooled, Wf0, bf0,
                                                                           zbuf, G, HID, 1);
    gemm_strip_wmma_f32<4><<<dim3((fStrips + 3) / 4), 128, shB4, stream>>>(zbuf, Wf1, bf1,
                                                                           out, G, HID, 0);
}